// MZIB_6047313953028
// MI455X (gfx1250) — compile-verified
//
#include <hip/hip_runtime.h>
#include <hip/hip_bf16.h>

// MZI butterfly rotation, (8193 x 8192) fp32, pure streaming kernel.
// Roofline: ~537 MB of traffic, ~0.4 GFLOP -> HBM-bound, floor ~23 us @ 23.3 TB/s.
// Strategy: b128 vmem, non-temporal hints (stream > 192MB L2), and gfx1250
// async global->LDS double-buffered pipeline (cp.async analog) to hide HBM
// latency with modest occupancy. No WMMA: there is no reduction dimension.

typedef float f4 __attribute__((ext_vector_type(4)));

#define NFEAT 8193
#define BATCH 8192
#define HALF  4096
#define TPB   256
#define ITEMS 4          // float4 chunks per thread per row

#if defined(__gfx1250__) && \
    __has_builtin(__builtin_amdgcn_global_load_async_to_lds_b128) && \
    __has_builtin(__builtin_amdgcn_s_wait_asynccnt)
#define USE_ASYNC 1
#else
#define USE_ASYNC 0
#endif

#if USE_ASYNC
// Types matched to the builtin's signature from clang's diagnostic:
// param 1: 'int __attribute__((vector_size(16))) __device__ *'  (addrspace(1))
// param 2: LDS destination -> addrspace(3) pointer of the same vector type.
typedef int i4v __attribute__((vector_size(4 * sizeof(int))));
typedef __attribute__((address_space(1))) i4v gi4;
typedef __attribute__((address_space(3))) i4v li4;
#endif

__device__ __forceinline__ void rot2_store(f4 xv, f4 yv,
                                           float c0, float s0, float c1, float s1,
                                           f4* __restrict__ ox, f4* __restrict__ oy)
{
    f4 x1, y1, x2, y2;
#pragma unroll
    for (int j = 0; j < 4; ++j) {
        x1[j] = c0 * xv[j] - s0 * yv[j];
        y1[j] = s0 * xv[j] + c0 * yv[j];
        x2[j] = c1 * x1[j] - s1 * y1[j];
        y2[j] = s1 * x1[j] + c1 * y1[j];
    }
    __builtin_nontemporal_store(x2, ox);
    __builtin_nontemporal_store(y2, oy);
}

__global__ __launch_bounds__(TPB) void mzib_rot_kernel(
    const float* __restrict__ cmplx,
    const float* __restrict__ angleB0,
    const float* __restrict__ angleB1,
    float* __restrict__ out)
{
    const int pair = blockIdx.y;         // 0..HALF-1, uniform per block
    const int tid  = threadIdx.x;

    // Uniform per block -> scalar loads; sincos once per thread, negligible.
    const float a0 = angleB0[pair];
    const float a1 = angleB1[pair];
    float s0, c0, s1, c1;
    sincosf(a0, &s0, &c0);
    sincosf(a1, &s1, &c1);

    const size_t xoff = (size_t)(2 * pair + 1) * BATCH;   // row 2i+1
    const f4* __restrict__ xr = (const f4*)(cmplx + xoff);
    const f4* __restrict__ yr = xr + (BATCH / 4);          // row 2i+2
    f4* __restrict__ ox = (f4*)(out + xoff);
    f4* __restrict__ oy = ox + (BATCH / 4);

    const int base = blockIdx.x * (TPB * ITEMS) + tid;     // float4 index

#if USE_ASYNC
    // 2-stage software pipeline: async DMA global->LDS, each lane reads back
    // exactly the 16B it requested after s_wait_asynccnt (no barrier needed).
    __shared__ f4 sx[2][TPB];
    __shared__ f4 sy[2][TPB];

    __builtin_amdgcn_global_load_async_to_lds_b128(
        (gi4*)(xr + base), (li4*)&sx[0][tid], 0, 0);
    __builtin_amdgcn_global_load_async_to_lds_b128(
        (gi4*)(yr + base), (li4*)&sy[0][tid], 0, 0);

#pragma unroll
    for (int k = 0; k < ITEMS; ++k) {
        if (k + 1 < ITEMS) {
            const int nidx = base + (k + 1) * TPB;
            const int st   = (k + 1) & 1;
            __builtin_amdgcn_global_load_async_to_lds_b128(
                (gi4*)(xr + nidx), (li4*)&sx[st][tid], 0, 0);
            __builtin_amdgcn_global_load_async_to_lds_b128(
                (gi4*)(yr + nidx), (li4*)&sy[st][tid], 0, 0);
            __builtin_amdgcn_s_wait_asynccnt(2);   // stage k's 2 loads done
        } else {
            __builtin_amdgcn_s_wait_asynccnt(0);
        }
        const int idx = base + k * TPB;
        rot2_store(sx[k & 1][tid], sy[k & 1][tid], c0, s0, c1, s1,
                   ox + idx, oy + idx);
    }
#else
    // Fallback: fully unrolled independent NT b128 loads (8 loads in flight
    // per thread), NT stores.
#pragma unroll
    for (int k = 0; k < ITEMS; ++k) {
        const int idx = base + k * TPB;
        f4 xv = __builtin_nontemporal_load(xr + idx);
        f4 yv = __builtin_nontemporal_load(yr + idx);
        rot2_store(xv, yv, c0, s0, c1, s1, ox + idx, oy + idx);
    }
#endif
}

// Row 0 is the only pass-through row (the reference's out[-1]=cmplx[-1] is
// overwritten by the out[2::2]=y2 slice, which ends at row 8192).
__global__ __launch_bounds__(TPB) void mzib_edge_kernel(
    const float* __restrict__ cmplx, float* __restrict__ out)
{
    const int i = blockIdx.x * blockDim.x + threadIdx.x;   // float4 index
    if (i < BATCH / 4) {
        const f4* __restrict__ src = (const f4*)cmplx;
        f4* __restrict__ dst = (f4*)out;
        __builtin_nontemporal_store(__builtin_nontemporal_load(src + i), dst + i);
    }
}

extern "C" void kernel_launch(void* const* d_in, const int* in_sizes, int n_in,
                              void* d_out, int out_size, void* d_ws, size_t ws_size,
                              hipStream_t stream)
{
    const float* cmplx   = (const float*)d_in[0];
    const float* angleB0 = (const float*)d_in[1];
    const float* angleB1 = (const float*)d_in[2];
    float* out = (float*)d_out;

    dim3 grid(BATCH / (TPB * ITEMS * 4), HALF);   // (2, 4096)
    mzib_rot_kernel<<<grid, TPB, 0, stream>>>(cmplx, angleB0, angleB1, out);

    const int edge_blocks = (BATCH / 4 + TPB - 1) / TPB;  // 8
    mzib_edge_kernel<<<edge_blocks, TPB, 0, stream>>>(cmplx, out);
}